// ObservationPointNodeEmbedding_46033459478825
// MI455X (gfx1250) — compile-verified
//
#include <hip/hip_runtime.h>
#include <hip/hip_bf16.h>

#define N_NODES 50000
#define E_EDGES 800000
#define NEG_SLOPE 0.2f

typedef __attribute__((ext_vector_type(2))) float v2f;
typedef __attribute__((ext_vector_type(8))) float v8f;

// ---- order-preserving float <-> uint encoding for atomic max --------------
__device__ __forceinline__ unsigned flipf(float f) {
    unsigned u = __float_as_uint(f);
    return (u & 0x80000000u) ? ~u : (u | 0x80000000u);
}
__device__ __forceinline__ float unflipf(unsigned e) {
    unsigned u = (e & 0x80000000u) ? (e & 0x7FFFFFFFu) : ~e;
    return __uint_as_float(u);
}

// ---- fp32 WMMA GEMM: C[M,Kn] = A[M,K] * B[K,Kn] ---------------------------
// One wave32 per 16x16 output tile; V_WMMA_F32_16X16X4_F32, K fully unrolled.
// K and Kn are compile-time so every global load in the hot loop uses an
// immediate byte offset (no 64-bit address math per iteration).
// A 16x4 layout: lanes 0-15 hold K={0,1}, lanes 16-31 hold K={2,3}.
// B 4x16 / C 16x16: rows striped across lanes within a VGPR.
template <int K, int Kn>
__global__ __launch_bounds__(32) void gemm_wmma_f32(
    const float* __restrict__ A, const float* __restrict__ B,
    float* __restrict__ C)
{
    const int tm   = blockIdx.x;          // row tile
    const int tn   = blockIdx.y;          // col tile
    const int lane = threadIdx.x & 31;
    const int half = lane >> 4;
    const int l16  = lane & 15;

    const float* Arow = A + (size_t)(tm * 16 + l16) * K + half * 2;
    const int    coln = tn * 16 + l16;
    const float* Bcol = B + coln + (size_t)half * 2 * Kn;

    v8f acc = {};
#pragma unroll
    for (int k0 = 0; k0 < K; k0 += 4) {
        v2f a = *(const v2f*)(Arow + k0);          // A[row][k0+2h .. +1], 8B load
        v2f b;
        b.x = Bcol[(size_t)k0 * Kn];               // B[k0+2h  ][coln]
        b.y = Bcol[(size_t)(k0 + 1) * Kn];         // B[k0+2h+1][coln]
        acc = __builtin_amdgcn_wmma_f32_16x16x4_f32(
            false, a, false, b, (short)0, acc, false, false);
    }

    float* Crow = C + (size_t)(tm * 16 + half * 8) * Kn + coln;
#pragma unroll
    for (int r = 0; r < 8; ++r)
        Crow[(size_t)r * Kn] = acc[r];
}

// ---- out[n, HF] = bias[HF] (broadcast init; scatter adds on top) ----------
__global__ void init_out_bias(float* __restrict__ out, const float* __restrict__ bias,
                              int n, int HF)
{
    int i = blockIdx.x * blockDim.x + threadIdx.x;
    if (i < n * HF) out[i] = bias[i % HF];
}

__global__ void init_stats(unsigned* __restrict__ amax, float* __restrict__ denom, int n2)
{
    int i = blockIdx.x * blockDim.x + threadIdx.x;
    if (i < n2) { amax[i] = 0x007FFFFFu; /* flip(-inf) */ denom[i] = 0.0f; }
}

// ---- per-(node,head) attention dot products -------------------------------
__global__ void attn_dot_kernel(const float* __restrict__ xl,
                                const float* __restrict__ att_s,
                                const float* __restrict__ att_d,
                                float* __restrict__ a_s, float* __restrict__ a_d,
                                int n, int F)
{
    int i = blockIdx.x * blockDim.x + threadIdx.x;   // over n*2
    if (i >= n * 2) return;
    int node = i >> 1, h = i & 1;
    const float* row = xl + (size_t)node * (2 * F) + h * F;
    const float* as  = att_s + h * F;
    const float* ad  = att_d + h * F;
    float s = 0.f, t = 0.f;
    for (int f = 0; f < F; ++f) {
        float v = row[f];
        s += v * as[f];
        t += v * ad[f];
    }
    a_s[i] = s;
    a_d[i] = t;
}

__device__ __forceinline__ void edge_endpoints(int e, int E, const int* src,
                                               const int* dst, int& s, int& d)
{
    if (e < E) { s = src[e]; d = dst[e]; }
    else       { s = e - E; d = s; }          // self-loop
}

// ---- pass 1: segment max of leaky_relu(a_src[s]+a_dst[d]) over dst --------
__global__ void edge_max_kernel(const int* __restrict__ src, const int* __restrict__ dst,
                                const float* __restrict__ a_s, const float* __restrict__ a_d,
                                unsigned* __restrict__ amax, int E, int n)
{
    int e = blockIdx.x * blockDim.x + threadIdx.x;
    if (e >= E + n) return;
    int s, d;
    edge_endpoints(e, E, src, dst, s, d);
#pragma unroll
    for (int h = 0; h < 2; ++h) {
        float raw = a_s[s * 2 + h] + a_d[d * 2 + h];
        raw = raw > 0.f ? raw : NEG_SLOPE * raw;
        atomicMax(&amax[d * 2 + h], flipf(raw));
    }
}

// ---- pass 2: alpha = exp(raw - max); denom[d] += alpha --------------------
__global__ void edge_exp_kernel(const int* __restrict__ src, const int* __restrict__ dst,
                                const float* __restrict__ a_s, const float* __restrict__ a_d,
                                const unsigned* __restrict__ amax,
                                float* __restrict__ alpha, float* __restrict__ denom,
                                int E, int n)
{
    int e = blockIdx.x * blockDim.x + threadIdx.x;
    if (e >= E + n) return;
    int s, d;
    edge_endpoints(e, E, src, dst, s, d);
#pragma unroll
    for (int h = 0; h < 2; ++h) {
        float raw = a_s[s * 2 + h] + a_d[d * 2 + h];
        raw = raw > 0.f ? raw : NEG_SLOPE * raw;
        float m = unflipf(amax[d * 2 + h]);
        float a = __expf(raw - m);
        alpha[(size_t)e * 2 + h] = a;
        atomicAdd(&denom[d * 2 + h], a);
    }
}

// ---- pass 3: alpha /= denom[dst] ------------------------------------------
__global__ void edge_norm_kernel(const int* __restrict__ dst,
                                 float* __restrict__ alpha,
                                 const float* __restrict__ denom, int E, int n)
{
    int e = blockIdx.x * blockDim.x + threadIdx.x;
    if (e >= E + n) return;
    int d = (e < E) ? dst[e] : (e - E);
    alpha[(size_t)e * 2 + 0] *= __frcp_rn(denom[d * 2 + 0]);
    alpha[(size_t)e * 2 + 1] *= __frcp_rn(denom[d * 2 + 1]);
}

// ---- pass 4: out[dst] += alpha * xl[src]  (block = one edge, thread = feat)
__global__ void scatter_kernel(const float* __restrict__ xl,
                               const float* __restrict__ alpha,
                               const int* __restrict__ src, const int* __restrict__ dst,
                               float* __restrict__ out, int E, int F /* per-head */)
{
    int e  = blockIdx.x;
    int fi = threadIdx.x;
    int HF = blockDim.x;
    int s, d;
    edge_endpoints(e, E, src, dst, s, d);
    int h = (fi >= F) ? 1 : 0;
    float w = alpha[(size_t)e * 2 + h];
    atomicAdd(&out[(size_t)d * HF + fi], xl[(size_t)s * HF + fi] * w);
}

extern "C" void kernel_launch(void* const* d_in, const int* in_sizes, int n_in,
                              void* d_out, int out_size, void* d_ws, size_t ws_size,
                              hipStream_t stream)
{
    const float* x   = (const float*)d_in[0];
    const int*   eu  = (const int*)d_in[1];   // [2, E] undirected
    const int*   ed  = (const int*)d_in[2];   // [2, E] directed
    const float* W1  = (const float*)d_in[3];
    const float* as1 = (const float*)d_in[4];
    const float* ad1 = (const float*)d_in[5];
    const float* b1  = (const float*)d_in[6];
    const float* W2  = (const float*)d_in[7];
    const float* as2 = (const float*)d_in[8];
    const float* ad2 = (const float*)d_in[9];
    const float* b2  = (const float*)d_in[10];
    float* out = (float*)d_out;

    const int n = N_NODES, E = E_EDGES, Etot = E + n;

    // workspace partition (floats)
    float*    xl1   = (float*)d_ws;                    // N*128
    float*    h1    = xl1   + (size_t)n * 128;         // N*128 (layer-1 out)
    float*    xl2   = h1    + (size_t)n * 128;         // N*256
    float*    a_s   = xl2   + (size_t)n * 256;         // N*2
    float*    a_d   = a_s   + (size_t)n * 2;           // N*2
    float*    denom = a_d   + (size_t)n * 2;           // N*2
    unsigned* amax  = (unsigned*)(denom + (size_t)n * 2); // N*2
    float*    alpha = (float*)(amax + (size_t)n * 2);  // (E+N)*2

    const int *srcU = eu, *dstU = eu + E;
    const int *srcD = ed, *dstD = ed + E;

    const dim3 b256(256);
    const int  gE  = (Etot + 255) / 256;
    const int  gN2 = (n * 2 + 255) / 256;

    // ======================= layer 1 (F=64, HF=128) ========================
    init_out_bias<<<(n * 128 + 255) / 256, b256, 0, stream>>>(h1, b1, n, 128);
    init_stats<<<gN2, b256, 0, stream>>>(amax, denom, n * 2);
    gemm_wmma_f32<64, 128><<<dim3(n / 16, 128 / 16), dim3(32), 0, stream>>>(x, W1, xl1);
    attn_dot_kernel<<<gN2, b256, 0, stream>>>(xl1, as1, ad1, a_s, a_d, n, 64);
    edge_max_kernel<<<gE, b256, 0, stream>>>(srcU, dstU, a_s, a_d, amax, E, n);
    edge_exp_kernel<<<gE, b256, 0, stream>>>(srcU, dstU, a_s, a_d, amax, alpha, denom, E, n);
    edge_norm_kernel<<<gE, b256, 0, stream>>>(dstU, alpha, denom, E, n);
    scatter_kernel<<<Etot, dim3(128), 0, stream>>>(xl1, alpha, srcU, dstU, h1, E, 64);

    // ======================= layer 2 (F=128, HF=256) =======================
    init_out_bias<<<(n * 256 + 255) / 256, b256, 0, stream>>>(out, b2, n, 256);
    init_stats<<<gN2, b256, 0, stream>>>(amax, denom, n * 2);
    gemm_wmma_f32<128, 256><<<dim3(n / 16, 256 / 16), dim3(32), 0, stream>>>(h1, W2, xl2);
    attn_dot_kernel<<<gN2, b256, 0, stream>>>(xl2, as2, ad2, a_s, a_d, n, 128);
    edge_max_kernel<<<gE, b256, 0, stream>>>(srcD, dstD, a_s, a_d, amax, E, n);
    edge_exp_kernel<<<gE, b256, 0, stream>>>(srcD, dstD, a_s, a_d, amax, alpha, denom, E, n);
    edge_norm_kernel<<<gE, b256, 0, stream>>>(dstD, alpha, denom, E, n);
    scatter_kernel<<<Etot, dim3(256), 0, stream>>>(xl2, alpha, srcD, dstD, out, E, 128);
}